// Quadratic_76261439308486
// MI455X (gfx1250) — compile-verified
//
#include <hip/hip_runtime.h>

typedef __attribute__((ext_vector_type(2))) float v2f;
typedef __attribute__((ext_vector_type(8))) float v8f;

#define BDIM 128      // 4 waves of 32
#define MT   64       // batch rows per block (16 per wave)
#define D    128
#define DP1  129
#define OUTN 32
#define XSTR 132      // stride%64==4 -> conflict-free A-frag / xe reads
#define WSTR 132      // stride%64==4 -> conflict-free B-frag reads & stage writes
#define YSTR 33
#define NT   256      // 32 outputs * 8 j-tiles (j = 0..127; ones-column split out)

#if __has_builtin(__builtin_amdgcn_global_load_async_to_lds_b32)
#define ASYNC_LDS 1
typedef __attribute__((address_space(1))) int* gas1_t;   // prints as __device__ int*
typedef __attribute__((address_space(3))) int* las3_t;   // prints as __shared__ int*
#else
#define ASYNC_LDS 0
#endif

__device__ __forceinline__ v8f wmma_f32(v2f a, v2f b, v8f c) {
  return __builtin_amdgcn_wmma_f32_16x16x4_f32(false, a, false, b, (short)0, c,
                                               false, false);
}

__global__ __launch_bounds__(BDIM) void quad_wmma_kernel(
    const float* __restrict__ X, const float* __restrict__ Wg,
    const float* __restrict__ bias, float* __restrict__ Y) {
  __shared__ float xe[MT * XSTR];        // x tile [64][128]
  __shared__ float wbuf[2][16 * WSTR];   // double-buffered W tile (transposed)
  __shared__ float ylds[MT * YSTR];      // phase-1 partial results [64][32]

  const int tid  = threadIdx.x;
  const int lane = tid & 31;
  const int wave = tid >> 5;     // M-subtile owner
  const int h    = lane >> 4;    // lane half per ISA frag layouts
  const int jl   = lane & 15;
  const int mbase = blockIdx.x * MT;

  // staging thread mapping (128 threads x 16 elements = 16x128 tile)
  const int jl2 = tid & 15;      // column within tile
  const int kb  = tid >> 4;      // 0..7; element k = kb + 8u

#if ASYNC_LDS
#define ISSUE_TILE(tt)                                                        \
  do {                                                                        \
    const int o_ = (tt) >> 3, jt_ = (tt) & 7;                                 \
    const float* g_ = Wg + o_ * (D * DP1) + jt_ * 16 + jl2 + kb * DP1;        \
    float* l_ = &wbuf[(tt) & 1][jl2 * WSTR + kb];                             \
    _Pragma("unroll") for (int u_ = 0; u_ < 16; ++u_)                         \
        __builtin_amdgcn_global_load_async_to_lds_b32(                        \
            (gas1_t)(void*)(g_ + u_ * 8 * DP1),                               \
            (las3_t)(void*)(l_ + u_ * 8), 0, 0);                              \
  } while (0)
#if __has_builtin(__builtin_amdgcn_s_wait_asynccnt)
#define FINISH_TILE(tt) __builtin_amdgcn_s_wait_asynccnt(0)
#else
#define FINISH_TILE(tt) asm volatile("s_wait_asynccnt 0" ::: "memory")
#endif
#else
  float stg[16];
#define ISSUE_TILE(tt)                                                        \
  do {                                                                        \
    const int o_ = (tt) >> 3, jt_ = (tt) & 7;                                 \
    const float* g_ = Wg + o_ * (D * DP1) + jt_ * 16 + jl2 + kb * DP1;        \
    _Pragma("unroll") for (int u_ = 0; u_ < 16; ++u_)                         \
        stg[u_] = g_[u_ * 8 * DP1];                                           \
  } while (0)
#define FINISH_TILE(tt)                                                       \
  do {                                                                        \
    float* l_ = &wbuf[(tt) & 1][jl2 * WSTR + kb];                             \
    _Pragma("unroll") for (int u_ = 0; u_ < 16; ++u_) l_[u_ * 8] = stg[u_];   \
  } while (0)
#endif

  // ---- stage x tile into LDS ----
  for (int idx = tid; idx < MT * D; idx += BDIM) {
    int row = idx >> 7, col = idx & 127;
    xe[row * XSTR + col] = X[(mbase + row) * D + col];
  }
  __syncthreads();

  // ---- A fragments in registers (ISA 16x4 f32 layout per k-step) ----
  v2f a[32];
  {
    const float* arow = &xe[(wave * 16 + jl) * XSTR + 2 * h];
#pragma unroll
    for (int t = 0; t < 32; ++t) a[t] = *(const v2f*)&arow[4 * t];
  }

  // ---- phase 1: 256 tiles, double-buffered, async prefetch ----
  ISSUE_TILE(0);
  FINISH_TILE(0);
  __syncthreads();

  float racc[8];
#pragma unroll 2
  for (int idx = 0; idx < NT; ++idx) {
    const int jt = idx & 7;
    const int o  = idx >> 3;
    if (jt == 0) {
#pragma unroll
      for (int r = 0; r < 8; ++r) racc[r] = 0.0f;
    }
    if (idx + 1 < NT) ISSUE_TILE(idx + 1);   // prefetch into other buffer

    v8f c0 = {}; v8f c1 = {};
    const float* brow = &wbuf[idx & 1][jl * WSTR + 2 * h];
#pragma unroll
    for (int t = 0; t < 16; ++t) {
      v2f b0 = *(const v2f*)&brow[4 * t];
      v2f b1 = *(const v2f*)&brow[4 * (t + 16)];
      c0 = wmma_f32(a[t],      b0, c0);
      c1 = wmma_f32(a[t + 16], b1, c1);
    }

    // xe-weighted j reduction: c[r] = S[m = r+8h][j = jt*16 + jl]
#pragma unroll
    for (int r = 0; r < 8; ++r) {
      float xw = xe[(wave * 16 + r + 8 * h) * XSTR + jt * 16 + jl];
      racc[r] += (c0[r] + c1[r]) * xw;
    }

    if (jt == 7) {  // finished this output: reduce across 16-lane halves
#pragma unroll
      for (int r = 0; r < 8; ++r) {
        float v = racc[r];
        v += __shfl_xor(v, 1, 32);
        v += __shfl_xor(v, 2, 32);
        v += __shfl_xor(v, 4, 32);
        v += __shfl_xor(v, 8, 32);
        if (jl == 0) ylds[(wave * 16 + r + 8 * h) * YSTR + o] = v;
      }
    }

    if (idx + 1 < NT) FINISH_TILE(idx + 1);
    __syncthreads();
  }

  // ---- phase 2: ones-column term T2 = X @ V2, V2[k,o] = W[o, k*129 + 128] ----
  for (int ot = 0; ot < 2; ++ot) {
    __syncthreads();
    {
      // stage: wbuf0[o'][k] = W[(ot*16+o')*16512 + k*129 + 128]
      const float* wrow = Wg + (ot * 16 + jl2) * (D * DP1) + D + kb * DP1;
      float* l = &wbuf[0][jl2 * WSTR + kb];
#pragma unroll
      for (int u = 0; u < 16; ++u) l[u * 8] = wrow[u * 8 * DP1];
    }
    __syncthreads();

    v8f c0 = {}; v8f c1 = {};
    const float* brow = &wbuf[0][jl * WSTR + 2 * h];
#pragma unroll
    for (int t = 0; t < 16; ++t) {
      v2f b0 = *(const v2f*)&brow[4 * t];
      v2f b1 = *(const v2f*)&brow[4 * (t + 16)];
      c0 = wmma_f32(a[t],      b0, c0);
      c1 = wmma_f32(a[t + 16], b1, c1);
    }

    // C layout maps (m, o) directly: c[r] = T2[m = r+8h][o = ot*16 + jl]
    const float bo = bias[ot * 16 + jl];
#pragma unroll
    for (int r = 0; r < 8; ++r) {
      const int mrow = wave * 16 + r + 8 * h;
      float v = c0[r] + c1[r] + ylds[mrow * YSTR + ot * 16 + jl] + bo;
      Y[(mbase + mrow) * OUTN + ot * 16 + jl] = v;   // coalesced 64B chunks
    }
    __syncthreads();
  }
}

extern "C" void kernel_launch(void* const* d_in, const int* in_sizes, int n_in,
                              void* d_out, int out_size, void* d_ws, size_t ws_size,
                              hipStream_t stream) {
  const float* X  = (const float*)d_in[0];
  const float* Wg = (const float*)d_in[1];
  const float* b  = (const float*)d_in[2];
  float* Y = (float*)d_out;
  quad_wmma_kernel<<<dim3(16384 / MT), dim3(BDIM), 0, stream>>>(X, Wg, b, Y);
}